// MessagePhase_9337258901623
// MI455X (gfx1250) — compile-verified
//
#include <hip/hip_runtime.h>

// Problem constants (match reference)
#define DD 75      // atom feature dim
#define EE 6       // bond feature dim
#define TT 3       // message passing steps
#define NN 20000   // atoms  (20000 = 1250 * 16)
#define MM 40000   // edges
#define TD 225     // 3*D
#define TDP 240    // 3*D padded to multiple of 16
#define DP 76      // D padded to multiple of 4 (19 K-chunks of 4, 16B aligned rows)

typedef __attribute__((ext_vector_type(2))) float v2f;
typedef __attribute__((ext_vector_type(8))) float v8f;

__device__ __forceinline__ float elu_f(float x) {
    return x > 0.0f ? x : (__expf(x) - 1.0f);
}
__device__ __forceinline__ float sigmoid_f(float x) {
    return 1.0f / (1.0f + __expf(-x));
}

// ---------------------------------------------------------------------------
// f1 = elu(efeat @ W1^T + b1), stored padded [M, 76] (col 75 = 0).
// ---------------------------------------------------------------------------
__global__ void f1_kernel(const float* __restrict__ efeat,
                          const float* __restrict__ W1,
                          const float* __restrict__ b1,
                          float* __restrict__ f1p) {
    int idx = blockIdx.x * blockDim.x + threadIdx.x;
    if (idx >= MM * DP) return;
    int m = idx / DP, d = idx % DP;
    float r = 0.0f;
    if (d < DD) {
        const float* e = efeat + (size_t)m * EE;
        const float* w = W1 + (size_t)d * EE;
        float acc = b1[d];
#pragma unroll
        for (int j = 0; j < EE; ++j) acc = fmaf(e[j], w[j], acc);
        r = elu_f(acc);
    }
    f1p[idx] = r;
}

// Pad h [N,75] -> [N,76] with zero in col 75.
__global__ void pad_h_kernel(const float* __restrict__ h, float* __restrict__ hp) {
    int idx = blockIdx.x * blockDim.x + threadIdx.x;
    if (idx >= NN * DP) return;
    int n = idx / DP, d = idx % DP;
    hp[idx] = (d < DD) ? h[(size_t)n * DD + d] : 0.0f;
}

// Build padded, transposed-access-friendly weights: wpad[(k*2+which)*240+c][76]
// wpad[...][c][d] = W[k][c][d] for c<225, d<75, else 0.
__global__ void pad_w_kernel(const float* __restrict__ Wih,
                             const float* __restrict__ Whh,
                             float* __restrict__ wpad) {
    int idx = blockIdx.x * blockDim.x + threadIdx.x;
    const int total = TT * 2 * TDP * DP;
    if (idx >= total) return;
    int d = idx % DP;
    int c = (idx / DP) % TDP;
    int which = (idx / (DP * TDP)) % 2;
    int k = idx / (DP * TDP * 2);
    float v = 0.0f;
    if (c < TD && d < DD) {
        const float* W = which ? Whh : Wih;
        v = W[((size_t)k * TD + c) * DD + d];
    }
    wpad[idx] = v;
}

__global__ void zero_kernel(float* __restrict__ p, int n) {
    int i = blockIdx.x * blockDim.x + threadIdx.x;
    if (i < n) p[i] = 0.0f;
}

__global__ void copy_kernel(float* __restrict__ out, const float* __restrict__ in, int n) {
    int i = blockIdx.x * blockDim.x + threadIdx.x;
    if (i < n) out[i] = in[i];
}

// ---------------------------------------------------------------------------
// Message phase: one wave32 per edge, A_e regenerated on the fly:
//   msg[m,e] = sum_d h[src[m],d] * elu(f1[m,d]*W2[e] + b2[e])
// Lanes own e = lane, lane+32, lane+64. Rows are padded (stride 76, col75=0)
// and 16B-aligned -> float4 loads. Scatter via f32 atomics into mbuf (stride 76).
// ---------------------------------------------------------------------------
__global__ void msg_kernel(const float* __restrict__ f1p,
                           const float* __restrict__ hp,     // padded [N,76]
                           const float* __restrict__ W2,
                           const float* __restrict__ b2,
                           const int*   __restrict__ src,
                           const int*   __restrict__ dst,
                           float* __restrict__ mbuf) {       // padded [N,76]
    int wave = (blockIdx.x * blockDim.x + threadIdx.x) >> 5;
    int lane = threadIdx.x & 31;
    if (wave >= MM) return;
    int s = src[wave], t = dst[wave];
    const float4* fr4 = (const float4*)(f1p + (size_t)wave * DP);
    const float4* hr4 = (const float4*)(hp + (size_t)s * DP);

    int e0 = lane, e1 = lane + 32, e2 = lane + 64;   // e0,e1 always < 75
    float w0 = W2[e0], c0 = b2[e0];
    float w1 = W2[e1], c1 = b2[e1];
    float w2 = (e2 < DD) ? W2[e2] : 0.0f;
    float c2 = (e2 < DD) ? b2[e2] : 0.0f;

    float a0 = 0.0f, a1 = 0.0f, a2 = 0.0f;
    for (int j = 0; j < DP / 4; ++j) {               // 19 iterations
        float4 f = fr4[j];
        float4 hv = hr4[j];
        // padded slot contributes 0 because h[...,75] == 0
        a0 = fmaf(hv.x, elu_f(fmaf(f.x, w0, c0)), a0);
        a1 = fmaf(hv.x, elu_f(fmaf(f.x, w1, c1)), a1);
        a2 = fmaf(hv.x, elu_f(fmaf(f.x, w2, c2)), a2);
        a0 = fmaf(hv.y, elu_f(fmaf(f.y, w0, c0)), a0);
        a1 = fmaf(hv.y, elu_f(fmaf(f.y, w1, c1)), a1);
        a2 = fmaf(hv.y, elu_f(fmaf(f.y, w2, c2)), a2);
        a0 = fmaf(hv.z, elu_f(fmaf(f.z, w0, c0)), a0);
        a1 = fmaf(hv.z, elu_f(fmaf(f.z, w1, c1)), a1);
        a2 = fmaf(hv.z, elu_f(fmaf(f.z, w2, c2)), a2);
        a0 = fmaf(hv.w, elu_f(fmaf(f.w, w0, c0)), a0);
        a1 = fmaf(hv.w, elu_f(fmaf(f.w, w1, c1)), a1);
        a2 = fmaf(hv.w, elu_f(fmaf(f.w, w2, c2)), a2);
    }
    float* outp = mbuf + (size_t)t * DP;
    atomicAdd(outp + e0, a0);
    atomicAdd(outp + e1, a1);
    if (e2 < DD) atomicAdd(outp + e2, a2);
}

// ---------------------------------------------------------------------------
// Fused GRU step via V_WMMA_F32_16X16X4_F32 (fp32, no precision loss).
// Block = 128 threads (4 waves) owns 16 atoms. A-tiles (m and h rows, padded
// stride 76) are staged once in LDS; B comes from the zero-padded weight
// buffer. All hot-loop loads are unconditional; control flow is wave-uniform
// so EXEC stays all-ones for WMMA.
// ---------------------------------------------------------------------------
__global__ __launch_bounds__(128)
void gru_kernel(const float* __restrict__ mbuf,   // padded [N,76]
                const float* __restrict__ hcur,   // padded [N,76]
                const float* __restrict__ WpI,    // [240,76] zero-padded
                const float* __restrict__ WpH,    // [240,76] zero-padded
                const float* __restrict__ bih,    // [3D]
                const float* __restrict__ bhh,    // [3D]
                float* __restrict__ hnext,        // stride = ostride
                int ostride) {
    __shared__ float sAm[16][DP];
    __shared__ float sAh[16][DP];
    __shared__ float gi_s[16][TDP];
    __shared__ float gh_s[16][TDP];

    const int n0   = blockIdx.x * 16;
    const int tid  = threadIdx.x;
    const int wv   = tid >> 5;
    const int lane = tid & 31;
    const int half = lane >> 4;     // 0: lanes 0-15, 1: lanes 16-31
    const int l15  = lane & 15;

    // Stage A rows (both gemms) into LDS: 16 rows x 76 cols each.
    for (int i = tid; i < 16 * DP; i += 128) {
        int r = i / DP, c = i % DP;
        sAm[r][c] = mbuf[(size_t)(n0 + r) * DP + c];
        sAh[r][c] = hcur[(size_t)(n0 + r) * DP + c];
    }
    __syncthreads();

    // 15 column tiles of the [16 x 240] outputs, round-robin over 4 waves.
    for (int ct = wv; ct < TDP / 16; ct += 4) {
        const int c = ct * 16 + l15;                  // this lane's output col
        const float* wirow = WpI + (size_t)c * DP;    // B[d,c] = Wpad[c][d]
        const float* whrow = WpH + (size_t)c * DP;

        v8f acc_i = {};
        v8f acc_h = {};
#pragma unroll
        for (int kk = 0; kk < DP; kk += 4) {          // 19 chunks, no remainder
            const int ka = kk + half * 2;             // K for this lane half
            v2f Ai = *(const v2f*)&sAm[l15][ka];
            v2f Ah = *(const v2f*)&sAh[l15][ka];
            v2f Bi = *(const v2f*)(wirow + ka);
            v2f Bh = *(const v2f*)(whrow + ka);
            acc_i = __builtin_amdgcn_wmma_f32_16x16x4_f32(
                false, Ai, false, Bi, (short)0, acc_i, false, false);
            acc_h = __builtin_amdgcn_wmma_f32_16x16x4_f32(
                false, Ah, false, Bh, (short)0, acc_h, false, false);
        }
        // D layout: VGPR r -> rows r (lanes 0-15) / r+8 (lanes 16-31), col l15
#pragma unroll
        for (int r = 0; r < 8; ++r) {
            const int row = r + half * 8;
            gi_s[row][ct * 16 + l15] = acc_i[r];
            gh_s[row][ct * 16 + l15] = acc_h[r];
        }
    }
    __syncthreads();

    // Gate phase: 16 atoms x 75 features over 128 threads.
    for (int idx = tid; idx < 16 * DD; idx += 128) {
        const int r = idx / DD;
        const int c = idx % DD;
        const int n = n0 + r;
        const float i_r = gi_s[r][c]          + bih[c];
        const float h_r = gh_s[r][c]          + bhh[c];
        const float i_z = gi_s[r][DD + c]     + bih[DD + c];
        const float h_z = gh_s[r][DD + c]     + bhh[DD + c];
        const float i_n = gi_s[r][2 * DD + c] + bih[2 * DD + c];
        const float h_n = gh_s[r][2 * DD + c] + bhh[2 * DD + c];
        const float rr = sigmoid_f(i_r + h_r);
        const float zz = sigmoid_f(i_z + h_z);
        const float ng = tanhf(fmaf(rr, h_n, i_n));
        const float ho = sAh[r][c];                        // h_old from LDS
        hnext[(size_t)n * ostride + c] = fmaf(zz, ho - ng, ng);
    }
}

// ---------------------------------------------------------------------------
extern "C" void kernel_launch(void* const* d_in, const int* in_sizes, int n_in,
                              void* d_out, int out_size, void* d_ws, size_t ws_size,
                              hipStream_t stream) {
    const float* h     = (const float*)d_in[0];
    const float* efeat = (const float*)d_in[1];
    const float* W1    = (const float*)d_in[2];
    const float* b1    = (const float*)d_in[3];
    const float* W2    = (const float*)d_in[4];
    const float* b2    = (const float*)d_in[5];
    const float* Wih   = (const float*)d_in[6];
    const float* Whh   = (const float*)d_in[7];
    const float* bih   = (const float*)d_in[8];
    const float* bhh   = (const float*)d_in[9];
    const int*   src   = (const int*)d_in[10];
    const int*   dst   = (const int*)d_in[11];
    float* out = (float*)d_out;                   // [h_final (N*75) | h0 (N*75)]

    float* f1p  = (float*)d_ws;                   // M*76
    float* mbuf = f1p  + (size_t)MM * DP;         // N*76
    float* hp0  = mbuf + (size_t)NN * DP;         // N*76 padded input h
    float* hA   = hp0  + (size_t)NN * DP;         // N*76
    float* hB   = hA   + (size_t)NN * DP;         // N*76
    float* wpad = hB   + (size_t)NN * DP;         // 3*2*240*76

    // h0 -> second half of output
    copy_kernel<<<(NN * DD + 255) / 256, 256, 0, stream>>>(out + (size_t)NN * DD, h, NN * DD);
    // prep: padded f1, padded h, padded weights, zero h double-buffers
    f1_kernel<<<(MM * DP + 255) / 256, 256, 0, stream>>>(efeat, W1, b1, f1p);
    pad_h_kernel<<<(NN * DP + 255) / 256, 256, 0, stream>>>(h, hp0);
    pad_w_kernel<<<(TT * 2 * TDP * DP + 255) / 256, 256, 0, stream>>>(Wih, Whh, wpad);
    zero_kernel<<<(2 * NN * DP + 255) / 256, 256, 0, stream>>>(hA, 2 * NN * DP);

    const float* hcur = hp0;
    for (int k = 0; k < TT; ++k) {
        float* hnxt  = (k == 0) ? hA : (k == 1) ? hB : out;
        int ostride  = (k == TT - 1) ? DD : DP;
        zero_kernel<<<(NN * DP + 255) / 256, 256, 0, stream>>>(mbuf, NN * DP);
        msg_kernel<<<MM / 8, 256, 0, stream>>>(f1p, hcur, W2, b2, src, dst, mbuf);
        gru_kernel<<<NN / 16, 128, 0, stream>>>(
            mbuf, hcur,
            wpad + (size_t)(k * 2 + 0) * TDP * DP,
            wpad + (size_t)(k * 2 + 1) * TDP * DP,
            bih + (size_t)k * TD,
            bhh + (size_t)k * TD,
            hnxt, ostride);
        hcur = hnxt;
    }
}